// Rayleigh_Layer_1_VertexUpdate_91096256348959
// MI455X (gfx1250) — compile-verified
//
#include <hip/hip_runtime.h>

// ---------------------------------------------------------------------------
// segment_sum scatter-add + finalize for:
//   cbar_i = segment_sum(edge_attr, dst=edgeij_pair[1], num_segments=NV)
//   out[i] = [b_i, b_i * cbar_i],  b_i = vertex_attr[i,0]
//
// Memory-bound: ~400 MB HBM/call -> ~17us roofline @ 23.3 TB/s.
// Edge streams use non-temporal B128 loads (keep 4MB accumulator in L2);
// scatter uses native non-returning global_atomic_add_f32 at DEV scope.
// edge_scatter_kernel placed first so the disasm snippet shows its codegen.
// ---------------------------------------------------------------------------

typedef __attribute__((ext_vector_type(4))) float     v4f;
typedef __attribute__((ext_vector_type(2))) float     v2f;
typedef long long                                     i64;
typedef __attribute__((ext_vector_type(2))) long long v2i64;

__device__ __forceinline__ void atomic_add_f32_noret(float* p, float v) {
#if defined(__gfx1250__)
    // Non-returning FP32 atomic: tracked by STOREcnt, no return traffic.
    // DEV scope: RMW resolves at the L2 atomic units (accumulator is
    // L2-resident: 4 MB << 192 MB).
    asm volatile("global_atomic_add_f32 %0, %1, off scope:SCOPE_DEV"
                 :: "v"(p), "v"(v) : "memory");
#else
    atomicAdd(p, v);
#endif
}

// Kernel: 8 edges per thread; NT streaming B128 loads + f32 atomic scatter.
// 4x global_load_b128 (8 int64 indices) + 2x global_load_b128 (8 f32 values)
// per thread -> long VMEM clauses feeding 8 non-returning atomics.
__global__ void edge_scatter_kernel(const i64* __restrict__ dst,
                                    const float* __restrict__ eattr,
                                    float* __restrict__ acc,
                                    int n_edges) {
    long base = (long)(blockIdx.x * blockDim.x + threadIdx.x) * 8;
    if (base + 7 < n_edges) {
        const v2i64* dp = (const v2i64*)(dst + base);
        v2i64 d01 = __builtin_nontemporal_load(dp + 0);
        v2i64 d23 = __builtin_nontemporal_load(dp + 1);
        v2i64 d45 = __builtin_nontemporal_load(dp + 2);
        v2i64 d67 = __builtin_nontemporal_load(dp + 3);
        const v4f* ep = (const v4f*)(eattr + base);
        v4f e0 = __builtin_nontemporal_load(ep + 0);
        v4f e1 = __builtin_nontemporal_load(ep + 1);
        atomic_add_f32_noret(acc + (int)d01.x, e0.x);
        atomic_add_f32_noret(acc + (int)d01.y, e0.y);
        atomic_add_f32_noret(acc + (int)d23.x, e0.z);
        atomic_add_f32_noret(acc + (int)d23.y, e0.w);
        atomic_add_f32_noret(acc + (int)d45.x, e1.x);
        atomic_add_f32_noret(acc + (int)d45.y, e1.y);
        atomic_add_f32_noret(acc + (int)d67.x, e1.z);
        atomic_add_f32_noret(acc + (int)d67.y, e1.w);
    } else {
#pragma clang loop unroll(disable)
        for (long i = base; i < n_edges; ++i)
            atomic_add_f32_noret(acc + (int)dst[i], eattr[i]);
    }
}

// Kernel: zero the ws accumulator (d_ws is poisoned 0xAA by the harness).
__global__ void zero_accum_kernel(float* __restrict__ acc, int n) {
    int i4 = (blockIdx.x * blockDim.x + threadIdx.x) * 4;
    if (i4 + 3 < n) {
        *(v4f*)(acc + i4) = (v4f){0.f, 0.f, 0.f, 0.f};
    } else {
#pragma clang loop unroll(disable)
        for (int i = i4; i < n; ++i) acc[i] = 0.f;
    }
}

// Kernel: out[i] = { b_i, b_i * cbar_i }  (vectorized float2 in/out).
__global__ void finalize_kernel(const float* __restrict__ vattr,
                                const float* __restrict__ acc,
                                float* __restrict__ out,
                                int n_vertices) {
    int i = blockIdx.x * blockDim.x + threadIdx.x;
    if (i < n_vertices) {
        v2f v = __builtin_nontemporal_load((const v2f*)vattr + i);
        float b = v.x;
        float c = acc[i];
        v2f o = {b, b * c};
        __builtin_nontemporal_store(o, (v2f*)out + i);
    }
}

extern "C" void kernel_launch(void* const* d_in, const int* in_sizes, int n_in,
                              void* d_out, int out_size, void* d_ws, size_t ws_size,
                              hipStream_t stream) {
    const float* vertex_attr = (const float*)d_in[0];          // (NV, 2) f32
    const i64*   edgeij      = (const i64*)d_in[1];            // (2, NE) i64
    const float* edge_attr   = (const float*)d_in[2];          // (NE, 1) f32
    // d_in[3] (g) and d_in[4] (batch) are unused by the reference math.

    const int n_edges    = in_sizes[2];
    const int n_vertices = in_sizes[0] / 2;
    const i64* dst = edgeij + n_edges;                         // row 1 = dst

    float* acc = (float*)d_ws;                                 // NV floats
    float* out = (float*)d_out;                                // (NV, 2) f32

    const int B = 256;

    // 1) zero accumulator
    {
        int threads = (n_vertices + 3) / 4;
        int grid = (threads + B - 1) / B;
        zero_accum_kernel<<<grid, B, 0, stream>>>(acc, n_vertices);
    }
    // 2) scatter-add edges (8 edges/thread)
    {
        int threads = (n_edges + 7) / 8;
        int grid = (threads + B - 1) / B;
        edge_scatter_kernel<<<grid, B, 0, stream>>>(dst, edge_attr, acc, n_edges);
    }
    // 3) finalize output rows
    {
        int grid = (n_vertices + B - 1) / B;
        finalize_kernel<<<grid, B, 0, stream>>>(vertex_attr, acc, out, n_vertices);
    }
}